// Asymm_3d_spconv_8211977470573
// MI455X (gfx1250) — compile-verified
//
#include <hip/hip_runtime.h>

// f32 WMMA fragment types: 16x16x4_f32 -> A/B are 2 floats/lane, C/D are 8 floats/lane (wave32)
typedef float v2f __attribute__((ext_vector_type(2)));
typedef float v8f __attribute__((ext_vector_type(8)));

// ---------------------------------------------------------------------------
// Submanifold conv as rulebook gather-GEMM with V_WMMA_F32_16X16X4_F32.
// One wave computes a 16-row x COUT tile:  D[16,COUT] = sum_k gather_k(X)[16,CIN] * W[k][CIN,COUT]
// A layout (16x4 f32): lane L (L<16): row=L, K={0,1}; lane L>=16: row=L-16, K={2,3}
// B layout (4x16 f32): lane L: col=L%16, rows K={2*(L/16), 2*(L/16)+1}
// D layout (16x16 f32): VGPR r, lane L: row = r + 8*(L/16), col = L%16
// Leaky-ReLU (slope 0.01) fused in the epilogue.
// ---------------------------------------------------------------------------
template<int CIN, int COUT>
__global__ __launch_bounds__(256) void conv_wmma(const float* __restrict__ X,
                                                 const int*   __restrict__ nbr,
                                                 const float* __restrict__ W,
                                                 float* __restrict__ Y,
                                                 int nrows)
{
    constexpr int NT = COUT / 16;              // column tiles
    const int lane    = threadIdx.x & 31;
    const int wave    = threadIdx.x >> 5;
    const int rowBase = (blockIdx.x * 8 + wave) * 16;
    if (rowBase >= nrows) return;              // uniform per wave: EXEC stays all-1s

    const int n16 = lane & 15;                 // row (for A/gather) and col (for B/D)
    const int h   = lane >> 4;                 // half select

    v8f acc[NT];
    #pragma unroll
    for (int t = 0; t < NT; ++t) {
        #pragma unroll
        for (int i = 0; i < 8; ++i) acc[t][i] = 0.0f;
    }

    for (int k = 0; k < 9; ++k) {
        const int  gi    = nbr[(long)(rowBase + n16) * 9 + k];
        const bool valid = (gi < nrows);                       // idx==N  =>  missing neighbor
        const float mask = valid ? 1.0f : 0.0f;                // branchless zero row
        const float* xrow = X + (long)(valid ? gi : 0) * CIN + 2 * h;
        const float* wk   = W + (long)k * CIN * COUT;
        #pragma unroll
        for (int j = 0; j < CIN / 4; ++j) {                    // K in steps of 4
            v2f a;
            a.x = xrow[4 * j]     * mask;
            a.y = xrow[4 * j + 1] * mask;
            const float* wp = wk + (long)(4 * j + 2 * h) * COUT + n16;
            #pragma unroll
            for (int t = 0; t < NT; ++t) {
                v2f b;
                b.x = wp[t * 16];
                b.y = wp[COUT + t * 16];
                acc[t] = __builtin_amdgcn_wmma_f32_16x16x4_f32(
                    false, a, false, b, (short)0, acc[t], false, false);
            }
        }
    }

    // epilogue: leaky relu + store
    #pragma unroll
    for (int t = 0; t < NT; ++t) {
        #pragma unroll
        for (int rr = 0; rr < 8; ++rr) {
            float v = acc[t][rr];
            v = (v > 0.0f) ? v : 0.01f * v;
            Y[(long)(rowBase + rr + 8 * h) * COUT + t * 16 + n16] = v;
        }
    }
}

// ---------------------------------------------------------------------------
// Per-channel sum / sum-of-squares for training-mode BatchNorm (biased var).
// LDS partials per block, then one global float atomic per channel per block.
// ---------------------------------------------------------------------------
template<int C>
__global__ __launch_bounds__(256) void stats_k(const float* __restrict__ Y,
                                               float* __restrict__ stats,  // [2*C]: sum, sumsq
                                               int nrows)
{
    __shared__ float sh[2 * C];
    if (threadIdx.x < 2 * C) sh[threadIdx.x] = 0.0f;
    __syncthreads();

    const int c   = threadIdx.x & (C - 1);
    const int r0  = threadIdx.x / C;
    const int RPB = 256 / C;
    float s = 0.0f, s2 = 0.0f;
    for (int n = blockIdx.x * RPB + r0; n < nrows; n += gridDim.x * RPB) {
        float v = Y[(long)n * C + c];
        s  += v;
        s2 += v * v;
    }
    atomicAdd(&sh[c], s);
    atomicAdd(&sh[C + c], s2);
    __syncthreads();
    if (threadIdx.x < 2 * C) atomicAdd(stats + threadIdx.x, sh[threadIdx.x]);
}

// ---------------------------------------------------------------------------
// BN apply: dst = (y - mean) * g / sqrt(var + eps) + b  (+ add, for residual)
// ---------------------------------------------------------------------------
template<int C>
__global__ __launch_bounds__(256) void bn_k(const float* __restrict__ Yin,
                                            const float* __restrict__ stats,
                                            const float* __restrict__ g,
                                            const float* __restrict__ b,
                                            const float* __restrict__ add,
                                            float* __restrict__ dst,
                                            int nrows, float invN)
{
    long i = (long)blockIdx.x * blockDim.x + threadIdx.x;
    if (i >= (long)nrows * C) return;
    int c = (int)(i & (C - 1));
    float m   = stats[c] * invN;
    float var = stats[C + c] * invN - m * m;
    float sc  = g[c] * rsqrtf(var + 1e-5f);
    float v   = (Yin[i] - m) * sc + b[c];
    if (add) v += add[i];
    dst[i] = v;
}

__global__ void zero_k(float* p, int n)
{
    int i = blockIdx.x * blockDim.x + threadIdx.x;
    if (i < n) p[i] = 0.0f;
}

// ---------------------------------------------------------------------------
extern "C" void kernel_launch(void* const* d_in, const int* in_sizes, int n_in,
                              void* d_out, int out_size, void* d_ws, size_t ws_size,
                              hipStream_t stream)
{
    (void)n_in; (void)out_size; (void)ws_size;

    const float* x      = (const float*)d_in[0];
    const int*   nbr133 = (const int*)  d_in[1];
    const int*   nbr313 = (const int*)  d_in[2];
    const float* W_c1   = (const float*)d_in[3];
    const float* g0     = (const float*)d_in[4];
    const float* b0     = (const float*)d_in[5];
    const float* W_c12  = (const float*)d_in[6];
    const float* g02    = (const float*)d_in[7];
    const float* b02    = (const float*)d_in[8];
    const float* W_c2   = (const float*)d_in[9];
    const float* W_c3   = (const float*)d_in[10];
    const float* g2     = (const float*)d_in[11];
    const float* b2     = (const float*)d_in[12];
    const float* W_r1   = (const float*)d_in[13];
    const float* rg0    = (const float*)d_in[14];
    const float* rb0    = (const float*)d_in[15];
    const float* W_r12  = (const float*)d_in[16];
    const float* rg02   = (const float*)d_in[17];
    const float* rb02   = (const float*)d_in[18];
    const float* W_r2   = (const float*)d_in[19];
    const float* rg1    = (const float*)d_in[20];
    const float* rb1    = (const float*)d_in[21];
    const float* W_r3   = (const float*)d_in[22];
    const float* rg2    = (const float*)d_in[23];
    const float* rb2    = (const float*)d_in[24];

    const int N = in_sizes[0] / 16;            // x is [N, 16]
    float* A     = (float*)d_ws;               // N*64 floats
    float* B     = A + (size_t)N * 64;         // N*64 floats
    float* C     = B + (size_t)N * 64;         // N*32 floats (holds y)
    float* stats = C + (size_t)N * 32;         // 128 floats
    float* out   = (float*)d_out;              // N*64 floats, also used as conv scratch
    const float invN = 1.0f / (float)N;

    const dim3 blk(256);
    const int convGrid = (N / 16 + 7) / 8;     // 8 waves (16-row tiles) per block
    const int bnGrid32 = (int)(((long)N * 32 + 255) / 256);
    const int bnGrid64 = (int)(((long)N * 64 + 255) / 256);

    // ---- ResContextBlock (16 -> 32) ----
    // s = bn(lrelu(subm(x, nbr133, W_c1)))
    conv_wmma<16, 32><<<convGrid, blk, 0, stream>>>(x, nbr133, W_c1, A, N);
    zero_k<<<1, 128, 0, stream>>>(stats, 64);
    stats_k<32><<<256, blk, 0, stream>>>(A, stats, N);
    bn_k<32><<<bnGrid32, blk, 0, stream>>>(A, stats, g0, b0, nullptr, A, N, invN);
    // s = bn(lrelu(subm(s, nbr313, W_c12)))
    conv_wmma<32, 32><<<convGrid, blk, 0, stream>>>(A, nbr313, W_c12, B, N);
    zero_k<<<1, 128, 0, stream>>>(stats, 64);
    stats_k<32><<<256, blk, 0, stream>>>(B, stats, N);
    bn_k<32><<<bnGrid32, blk, 0, stream>>>(B, stats, g02, b02, nullptr, B, N, invN);
    // r = lrelu(subm(x, nbr313, W_c2))   (bn skipped, faithful to reference)
    conv_wmma<16, 32><<<convGrid, blk, 0, stream>>>(x, nbr313, W_c2, A, N);
    // y = bn(lrelu(subm(r, nbr133, W_c3))) + s   -> C
    conv_wmma<32, 32><<<convGrid, blk, 0, stream>>>(A, nbr133, W_c3, C, N);
    zero_k<<<1, 128, 0, stream>>>(stats, 64);
    stats_k<32><<<256, blk, 0, stream>>>(C, stats, N);
    bn_k<32><<<bnGrid32, blk, 0, stream>>>(C, stats, g2, b2, B, C, N, invN);

    // ---- ResBlock (32 -> 64) ----
    // s2 = bn(lrelu(subm(y, nbr313, W_r1)))
    conv_wmma<32, 64><<<convGrid, blk, 0, stream>>>(C, nbr313, W_r1, A, N);
    zero_k<<<1, 128, 0, stream>>>(stats, 128);
    stats_k<64><<<256, blk, 0, stream>>>(A, stats, N);
    bn_k<64><<<bnGrid64, blk, 0, stream>>>(A, stats, rg0, rb0, nullptr, A, N, invN);
    // s2 = bn(lrelu(subm(s2, nbr133, W_r12)))
    conv_wmma<64, 64><<<convGrid, blk, 0, stream>>>(A, nbr133, W_r12, B, N);
    zero_k<<<1, 128, 0, stream>>>(stats, 128);
    stats_k<64><<<256, blk, 0, stream>>>(B, stats, N);
    bn_k<64><<<bnGrid64, blk, 0, stream>>>(B, stats, rg02, rb02, nullptr, B, N, invN);
    // r2 = bn(lrelu(subm(y, nbr133, W_r2)))
    conv_wmma<32, 64><<<convGrid, blk, 0, stream>>>(C, nbr133, W_r2, A, N);
    zero_k<<<1, 128, 0, stream>>>(stats, 128);
    stats_k<64><<<256, blk, 0, stream>>>(A, stats, N);
    bn_k<64><<<bnGrid64, blk, 0, stream>>>(A, stats, rg1, rb1, nullptr, A, N, invN);
    // out = bn(lrelu(subm(r2, nbr313, W_r3))) + s2
    conv_wmma<64, 64><<<convGrid, blk, 0, stream>>>(A, nbr313, W_r3, out, N);
    zero_k<<<1, 128, 0, stream>>>(stats, 128);
    stats_k<64><<<256, blk, 0, stream>>>(out, stats, N);
    bn_k<64><<<bnGrid64, blk, 0, stream>>>(out, stats, rg2, rb2, B, out, N, invN);
}